// TrajectoryGenerator_22849226015378
// MI455X (gfx1250) — compile-verified
//
#include <hip/hip_runtime.h>
#include <cmath>

namespace {

constexpr int NN = 64;          // nodes per graph
constexpr float EPS = 1e-5f;
constexpr float NEGV = -1e12f;
constexpr int LST = 65;         // LDS row stride (pad -> conflict-free columns)

typedef float v2f __attribute__((ext_vector_type(2)));
typedef float v8f __attribute__((ext_vector_type(8)));

struct Smem {
  float h[NN][LST];        // node features (input / normalized / layer-1 input)
  float hp[NN][LST];       // h_prime = h @ W
  float attn[2][NN][LST];  // attention matrices (two heads at a time)
  float w0l[NN][LST];      // W0eff[f][h*16+o] staged in LDS
  float w1l[NN][LST];      // W1[f][o] staged in LDS
  float red[4][NN];        // reduction scratch (norm partials / softmax max)
  float red2[4][NN];       // reduction scratch (norm sq / softmax sum)
  float mean[NN];
  float inv[NN];
  float sc[4][NN];         // per-head source scores
  float dc[4][NN];         // per-head dest scores
  float mask[NN];
  float cnt;
};

// Masked instance norm over nodes (axis=N) per channel, in-place on sm.h.
__device__ inline void instance_norm(Smem& sm, int tid) {
  const int c = tid & 63, grp = tid >> 6;
  float s = 0.f, q = 0.f;
  #pragma unroll
  for (int r = 0; r < 16; ++r) {
    int row = grp * 16 + r;
    float v = sm.h[row][c] * sm.mask[row];
    s += v;
    q += v * v;
  }
  sm.red[grp][c] = s;
  sm.red2[grp][c] = q;
  __syncthreads();
  if (tid < NN) {
    float ss = sm.red[0][tid] + sm.red[1][tid] + sm.red[2][tid] + sm.red[3][tid];
    float qq = sm.red2[0][tid] + sm.red2[1][tid] + sm.red2[2][tid] + sm.red2[3][tid];
    float cnt = sm.cnt;
    float mean = ss / cnt;
    float var = fmaxf(qq / cnt - mean * mean, 0.f);
    sm.mean[tid] = mean;
    sm.inv[tid] = rsqrtf(var + EPS);
  }
  __syncthreads();
  const int row = tid >> 2, base = (tid & 3) * 16;
  #pragma unroll
  for (int j = 0; j < 16; ++j) {
    int cc = base + j;
    sm.h[row][cc] = (sm.h[row][cc] - sm.mean[cc]) * sm.inv[cc];
  }
  __syncthreads();
}

// sm.hp[64x64] = sm.h[64x64] @ W[64x64], W staged in LDS. fp32 WMMA 16x16x4.
__device__ inline void matmul_hW(Smem& sm, int tid, const float (*wl)[LST]) {
  const int lane = tid & 31;
  const int wv = tid >> 5;
  const int lhalf = lane >> 4;   // 0: K pair {0,1}; 1: K pair {2,3}
  const int l16 = lane & 15;
  #pragma unroll
  for (int it = 0; it < 2; ++it) {
    const int tile = wv * 2 + it;
    const int ti = tile >> 2, tj = tile & 3;
    const int col = tj * 16 + l16;
    v8f acc = {};
    #pragma unroll 4
    for (int kk = 0; kk < 16; ++kk) {
      const int k0 = kk * 4 + lhalf * 2;
      v2f a, bf;
      a.x = sm.h[ti * 16 + l16][k0];
      a.y = sm.h[ti * 16 + l16][k0 + 1];
      bf.x = wl[k0][col];
      bf.y = wl[k0 + 1][col];
      acc = __builtin_amdgcn_wmma_f32_16x16x4_f32(false, a, false, bf,
                                                  (short)0, acc, false, false);
    }
    #pragma unroll
    for (int r = 0; r < 8; ++r)
      sm.hp[ti * 16 + r + 8 * lhalf][tj * 16 + l16] = acc[r];
  }
}

// Masked leaky-relu softmax for heads (hb, hb+1) -> sm.attn[0..1].
// 2 threads per (head,row), 32 columns each; register-cached values.
__device__ inline void softmax_pair(Smem& sm, int tid, int hb) {
  const int hl = tid >> 7;          // local head 0/1
  const int part = (tid >> 6) & 1;  // column half
  const int i = tid & 63;           // row
  const float si = sm.sc[hb + hl][i];
  const float* drow = sm.dc[hb + hl];
  const float mi = sm.mask[i];
  float vals[32];
  float mx = -3.4e38f;
  #pragma unroll
  for (int jj = 0; jj < 32; ++jj) {
    const int j = part * 32 + jj;
    float v = si + drow[j];
    v = fmaxf(v, 0.2f * v);                  // leaky relu, slope 0.2
    v = (mi * sm.mask[j] > 0.f) ? v : NEGV;  // adjacency mask
    vals[jj] = v;
    mx = fmaxf(mx, v);
  }
  sm.red[hl * 2 + part][i] = mx;
  __syncthreads();
  mx = fmaxf(sm.red[hl * 2][i], sm.red[hl * 2 + 1][i]);
  float sum = 0.f;
  #pragma unroll
  for (int jj = 0; jj < 32; ++jj) {
    vals[jj] = expf(vals[jj] - mx);
    sum += vals[jj];
  }
  sm.red2[hl * 2 + part][i] = sum;
  __syncthreads();
  sum = sm.red2[hl * 2][i] + sm.red2[hl * 2 + 1][i];
  const float rs = 1.f / sum;
  #pragma unroll
  for (int jj = 0; jj < 32; ++jj)
    sm.attn[hl][i][part * 32 + jj] = vals[jj] * rs;
  __syncthreads();
}

// Single-head softmax (layer 1) -> sm.attn[0]. 4 threads per row.
__device__ inline void softmax_single(Smem& sm, int tid) {
  const int part = tid >> 6;  // 0..3
  const int i = tid & 63;
  const float si = sm.sc[0][i];
  const float mi = sm.mask[i];
  float vals[16];
  float mx = -3.4e38f;
  #pragma unroll
  for (int jj = 0; jj < 16; ++jj) {
    const int j = part * 16 + jj;
    float v = si + sm.dc[0][j];
    v = fmaxf(v, 0.2f * v);
    v = (mi * sm.mask[j] > 0.f) ? v : NEGV;
    vals[jj] = v;
    mx = fmaxf(mx, v);
  }
  sm.red[part][i] = mx;
  __syncthreads();
  mx = fmaxf(fmaxf(sm.red[0][i], sm.red[1][i]),
             fmaxf(sm.red[2][i], sm.red[3][i]));
  float sum = 0.f;
  #pragma unroll
  for (int jj = 0; jj < 16; ++jj) {
    vals[jj] = expf(vals[jj] - mx);
    sum += vals[jj];
  }
  sm.red2[part][i] = sum;
  __syncthreads();
  sum = sm.red2[0][i] + sm.red2[1][i] + sm.red2[2][i] + sm.red2[3][i];
  const float rs = 1.f / sum;
  #pragma unroll
  for (int jj = 0; jj < 16; ++jj)
    sm.attn[0][i][part * 16 + jj] = vals[jj] * rs;
  __syncthreads();
}

__global__ __launch_bounds__(256) void gat_encoder_fused(
    const float* __restrict__ x, const int* __restrict__ fhi,
    const float* __restrict__ w0, const float* __restrict__ a_src0,
    const float* __restrict__ a_dst0, const float* __restrict__ b0,
    const float* __restrict__ w1, const float* __restrict__ a_src1,
    const float* __restrict__ a_dst1, const float* __restrict__ b1,
    float* __restrict__ out) {
  __shared__ Smem sm;
  const int tid = threadIdx.x;
  const int bg = blockIdx.x;      // graph index in [0, S*T)
  const int sIdx = bg >> 3;       // scene
  const int tIdx = bg & 7;        // timestep

  // Visibility mask: pedestrian n present iff fhi[s,n] <= t
  if (tid < NN) sm.mask[tid] = (fhi[sIdx * NN + tid] <= tIdx) ? 1.f : 0.f;

  // Stage x[bg] and both weight matrices into LDS (16 contiguous f32/thread).
  {
    const int row = tid >> 2, q = tid & 3;
    const float* xg = x + (size_t)bg * (NN * 64);
    const float4* sx = reinterpret_cast<const float4*>(xg + row * 64 + q * 16);
    // W0eff[f][h*16+o] = w0[h][f][o]; for fixed (f, h=q) the o-range is contiguous.
    const float4* s0 = reinterpret_cast<const float4*>(w0 + q * 1024 + row * 16);
    const float4* s1 = reinterpret_cast<const float4*>(w1 + row * 64 + q * 16);
    #pragma unroll
    for (int j = 0; j < 4; ++j) {
      float4 v = sx[j];
      sm.h[row][q * 16 + j * 4 + 0] = v.x;
      sm.h[row][q * 16 + j * 4 + 1] = v.y;
      sm.h[row][q * 16 + j * 4 + 2] = v.z;
      sm.h[row][q * 16 + j * 4 + 3] = v.w;
      float4 a = s0[j];
      sm.w0l[row][q * 16 + j * 4 + 0] = a.x;
      sm.w0l[row][q * 16 + j * 4 + 1] = a.y;
      sm.w0l[row][q * 16 + j * 4 + 2] = a.z;
      sm.w0l[row][q * 16 + j * 4 + 3] = a.w;
      float4 b = s1[j];
      sm.w1l[row][q * 16 + j * 4 + 0] = b.x;
      sm.w1l[row][q * 16 + j * 4 + 1] = b.y;
      sm.w1l[row][q * 16 + j * 4 + 2] = b.z;
      sm.w1l[row][q * 16 + j * 4 + 3] = b.w;
    }
  }
  __syncthreads();
  if (tid == 0) {
    float c = 0.f;
    for (int n = 0; n < NN; ++n) c += sm.mask[n];
    sm.cnt = fmaxf(c, 1.f);
  }
  __syncthreads();

  const int lane = tid & 31;
  const int wv = tid >> 5;
  const int lhalf = lane >> 4;
  const int l16 = lane & 15;

  // ===================== Layer 0 (4 heads, f_out = 16) =====================
  instance_norm(sm, tid);
  matmul_hW(sm, tid, sm.w0l);
  __syncthreads();

  // Attention scores for all 4 heads: one (head, node) per thread.
  {
    const int hh = tid >> 6, n = tid & 63;
    float ss = 0.f, dd = 0.f;
    #pragma unroll
    for (int o = 0; o < 16; ++o) {
      float hv = sm.hp[n][hh * 16 + o];
      ss += hv * a_src0[hh * 16 + o];
      dd += hv * a_dst0[hh * 16 + o];
    }
    sm.sc[hh][n] = ss;
    sm.dc[hh][n] = dd;
  }
  __syncthreads();

  // Two heads per pass: softmax, then out = attn @ h'[:, h*16:+16], +b0, ELU.
  const float bias0 = b0[l16];
  for (int hb = 0; hb < 4; hb += 2) {
    softmax_pair(sm, tid, hb);
    {
      const int hl = wv >> 2;     // local head handled by this wave
      const int ti = wv & 3;      // row tile
      const int hcol = (hb + hl) * 16 + l16;
      v8f acc = {};
      #pragma unroll 4
      for (int kk = 0; kk < 16; ++kk) {
        const int k0 = kk * 4 + lhalf * 2;
        v2f a, bf;
        a.x = sm.attn[hl][ti * 16 + l16][k0];
        a.y = sm.attn[hl][ti * 16 + l16][k0 + 1];
        bf.x = sm.hp[k0][hcol];
        bf.y = sm.hp[k0 + 1][hcol];
        acc = __builtin_amdgcn_wmma_f32_16x16x4_f32(false, a, false, bf,
                                                    (short)0, acc, false, false);
      }
      #pragma unroll
      for (int r = 0; r < 8; ++r) {
        const int row = ti * 16 + r + 8 * lhalf;
        float v = acc[r] + bias0;
        v = v > 0.f ? v : (expf(v) - 1.f);        // ELU (alpha = 1)
        sm.h[row][hcol] = v;                      // head concat (transpose)
      }
    }
    __syncthreads();
  }

  // ===================== Layer 1 (1 head, f_out = 64) =====================
  instance_norm(sm, tid);
  matmul_hW(sm, tid, sm.w1l);
  __syncthreads();

  if (tid < NN) {
    const int n = tid;
    float ss = 0.f, dd = 0.f;
    for (int o = 0; o < 64; ++o) {
      float hv = sm.hp[n][o];
      ss += hv * a_src1[o];
      dd += hv * a_dst1[o];
    }
    sm.sc[0][n] = ss;
    sm.dc[0][n] = dd;
  }
  __syncthreads();

  softmax_single(sm, tid);

  // out = attn @ h' + b1 : 16 tiles, 8 waves x 2, stored straight to global.
  float* og = out + (size_t)bg * (NN * 64);
  #pragma unroll
  for (int it = 0; it < 2; ++it) {
    const int tile = wv * 2 + it;
    const int ti = tile >> 2, tj = tile & 3;
    const float bias1 = b1[tj * 16 + l16];
    v8f acc = {};
    #pragma unroll 4
    for (int kk = 0; kk < 16; ++kk) {
      const int k0 = kk * 4 + lhalf * 2;
      v2f a, bf;
      a.x = sm.attn[0][ti * 16 + l16][k0];
      a.y = sm.attn[0][ti * 16 + l16][k0 + 1];
      bf.x = sm.hp[k0][tj * 16 + l16];
      bf.y = sm.hp[k0 + 1][tj * 16 + l16];
      acc = __builtin_amdgcn_wmma_f32_16x16x4_f32(false, a, false, bf,
                                                  (short)0, acc, false, false);
    }
    #pragma unroll
    for (int r = 0; r < 8; ++r) {
      const int row = ti * 16 + r + 8 * lhalf;
      const int col = tj * 16 + l16;
      og[row * 64 + col] = acc[r] + bias1;
    }
  }
}

}  // namespace

extern "C" void kernel_launch(void* const* d_in, const int* in_sizes, int n_in,
                              void* d_out, int out_size, void* d_ws,
                              size_t ws_size, hipStream_t stream) {
  (void)in_sizes; (void)n_in; (void)out_size; (void)d_ws; (void)ws_size;
  const float* x    = (const float*)d_in[0];
  const int*   fhi  = (const int*)d_in[1];
  const float* w0   = (const float*)d_in[2];
  const float* as0  = (const float*)d_in[3];
  const float* ad0  = (const float*)d_in[4];
  const float* b0   = (const float*)d_in[5];
  const float* w1   = (const float*)d_in[6];
  const float* as1  = (const float*)d_in[7];
  const float* ad1  = (const float*)d_in[8];
  const float* b1   = (const float*)d_in[9];
  float* out = (float*)d_out;

  dim3 grid(1024 * 8);   // S*T graphs
  dim3 block(256);       // 8 wave32 waves
  hipLaunchKernelGGL(gat_encoder_fused, grid, block, 0, stream,
                     x, fhi, w0, as0, ad0, b0, w1, as1, ad1, b1, out);
}